// AnchorManager_37529424232649
// MI455X (gfx1250) — compile-verified
//
#include <hip/hip_runtime.h>
#include <hip/hip_bf16.h>
#include <stdint.h>

#define B_SZ   32
#define N_GT   100
#define A_TOT  8732
#define A_TILES 546   // ceil(8732/16)
#define N_TILES 7     // ceil(100/16)
#define EPS_F  1e-6f

typedef float v2f __attribute__((ext_vector_type(2)));
typedef float v8f __attribute__((ext_vector_type(8)));

// ---------------------------------------------------------------------------
// Kernel 0: zero the packed (iou,anchor) argmax array (ws is poisoned 0xAA)
// ---------------------------------------------------------------------------
__global__ void init_packed_kernel(unsigned long long* __restrict__ packed, int count) {
  int i = blockIdx.x * blockDim.x + threadIdx.x;
  if (i < count) packed[i] = 0ULL;
}

// ---------------------------------------------------------------------------
// Kernel 1: one wave32 per (batch, 16-anchor tile).
//   - async-stage GT boxes (100 x 4 f32) into LDS  (ASYNCcnt path)
//   - per 16x16 (anchor x gt) tile: WMMA outer-sum area_a[m] + area_b[n]
//   - per-lane VALU: intersection, iou = inter / max(sum - inter, eps)
//   - row reduce  -> best_iou / best_idx per anchor
//   - col reduce  -> u64 atomicMax packed (iou_bits<<32 | ~m) per GT
// ---------------------------------------------------------------------------
__global__ __launch_bounds__(32) void iou_tile_kernel(
    const float* __restrict__ gt, const float* __restrict__ anchors,
    float* __restrict__ best_iou, int* __restrict__ best_idx,
    unsigned long long* __restrict__ packed)
{
  const int lane = threadIdx.x;               // 0..31 (wave32)
  const int tile = blockIdx.x % A_TILES;
  const int b    = blockIdx.x / A_TILES;
  const int base = tile * 16;

  __shared__ float sh_gt[N_GT * 4];
  __shared__ float sh_anc[16 * 4];

  // ---- async global->LDS staging of this batch's GT boxes (CDNA5) ----
  {
    const float* gsrc = gt + (size_t)b * N_GT * 4;
    unsigned ldsbase = (unsigned)(uintptr_t)(&sh_gt[0]);
    for (int i = 0; i < 4; ++i) {
      int idx = i * 32 + lane;
      if (idx < N_GT) {
        unsigned lds_addr = ldsbase + (unsigned)idx * 16u;
        unsigned long long gaddr = (unsigned long long)(uintptr_t)(gsrc + idx * 4);
        asm volatile("global_load_async_to_lds_b128 %0, %1, off"
                     :: "v"(lds_addr), "v"(gaddr) : "memory");
      }
    }
    asm volatile("s_wait_asynccnt 0" ::: "memory");
  }

  // ---- anchor tile: lanes 0..15 convert cxcywh -> xyxy into LDS ----
  float areaA = 0.0f;
  if (lane < 16) {
    int m = base + lane;
    float cx = 0.f, cy = 0.f, w = 0.f, h = 0.f;
    if (m < A_TOT) {
      const float* ap = anchors + (size_t)m * 4;
      cx = ap[0]; cy = ap[1]; w = ap[2]; h = ap[3];
    }
    float x1 = cx - 0.5f * w, y1 = cy - 0.5f * h;
    float x2 = cx + 0.5f * w, y2 = cy + 0.5f * h;
    sh_anc[lane * 4 + 0] = x1; sh_anc[lane * 4 + 1] = y1;
    sh_anc[lane * 4 + 2] = x2; sh_anc[lane * 4 + 3] = y2;
    areaA = (x2 - x1) * (y2 - y1);
  }
  __syncthreads();

  // D-matrix layout: VGPR r holds M=r (lanes 0-15) / M=r+8 (lanes 16-31), N=lane&15
  const int hiOff = (lane >> 4) * 8;
  float ax1[8], ay1[8], ax2[8], ay2[8];
  #pragma unroll
  for (int r = 0; r < 8; ++r) {
    int m = r + hiOff;
    ax1[r] = sh_anc[m * 4 + 0]; ay1[r] = sh_anc[m * 4 + 1];
    ax2[r] = sh_anc[m * 4 + 2]; ay2[r] = sh_anc[m * 4 + 3];
  }

  // A operand (16x4 f32): K0 = area_a[m], K1 = 1, K2 = K3 = 0 (upper half zeroed)
  v2f aop; aop.x = (lane < 16) ? areaA : 0.f; aop.y = (lane < 16) ? 1.f : 0.f;
  v8f cz = {0.f, 0.f, 0.f, 0.f, 0.f, 0.f, 0.f, 0.f};

  float runv[8]; int runn[8];
  #pragma unroll
  for (int r = 0; r < 8; ++r) { runv[r] = -2.0f; runn[r] = 0; }

  for (int t = 0; t < N_TILES; ++t) {
    int n = (lane & 15) + 16 * t;
    bool nvalid = n < N_GT;
    float gx1 = 0.f, gy1 = 0.f, gx2 = 0.f, gy2 = 0.f, areaB = 0.f;
    if (nvalid) {
      gx1 = sh_gt[n * 4 + 0]; gy1 = sh_gt[n * 4 + 1];
      gx2 = sh_gt[n * 4 + 2]; gy2 = sh_gt[n * 4 + 3];
      areaB = (gx2 - gx1) * (gy2 - gy1);   // unclipped, as in reference
    }
    // B operand (4x16 f32): K0 row = 1, K1 row = area_b[n]
    v2f bop; bop.x = (lane < 16) ? 1.f : 0.f; bop.y = (lane < 16) ? areaB : 0.f;

    // matrix core computes the bilinear part: d[m][n] = area_a[m] + area_b[n]
    v8f d = __builtin_amdgcn_wmma_f32_16x16x4_f32(
        /*neg_a=*/false, aop, /*neg_b=*/false, bop,
        /*c_mod=*/(short)0, cz, /*reuse_a=*/false, /*reuse_b=*/false);

    float colv = -2.0f; int colm = 0;
    #pragma unroll
    for (int r = 0; r < 8; ++r) {
      int m_glob = base + r + hiOff;
      float ltx = fmaxf(ax1[r], gx1), rbx = fminf(ax2[r], gx2);
      float lty = fmaxf(ay1[r], gy1), rby = fminf(ay2[r], gy2);
      float iw = fmaxf(rbx - ltx, 0.f), ih = fmaxf(rby - lty, 0.f);
      float inter = iw * ih;
      float uni = d[r] - inter;                 // area_a + area_b - inter
      float iou = inter / fmaxf(uni, EPS_F);
      if (m_glob >= A_TOT || !nvalid) iou = -1.f;
      // row running max (strict > keeps first/smallest n within this lane)
      if (iou > runv[r]) { runv[r] = iou; runn[r] = n; }
      // column max (ascending r keeps smallest m on ties via strict >,
      // explicit tie-break kept for the cross-half combine symmetry)
      if (iou > colv || (iou == colv && m_glob < colm)) { colv = iou; colm = m_glob; }
    }
    // combine column max across the two half-waves (lane L and L^16 share n)
    float ov = __shfl_xor(colv, 16, 32);
    int   om = __shfl_xor(colm, 16, 32);
    if (ov > colv || (ov == colv && om < colm)) { colv = ov; colm = om; }
    if (lane < 16 && nvalid && colv >= 0.f) {
      unsigned long long key =
          ((unsigned long long)__float_as_uint(colv) << 32) |
          (unsigned long long)(0xFFFFFFFFu - (unsigned)colm);
      atomicMax(&packed[(size_t)b * N_GT + n], key);
    }
  }

  // ---- row (per-anchor) reduction across the 16 lanes of each half ----
  #pragma unroll
  for (int r = 0; r < 8; ++r) {
    float v = runv[r]; int nn = runn[r];
    #pragma unroll
    for (int off = 1; off < 16; off <<= 1) {
      float v2 = __shfl_xor(v, off, 32);
      int   n2 = __shfl_xor(nn, off, 32);
      if (v2 > v || (v2 == v && n2 < nn)) { v = v2; nn = n2; }
    }
    if ((lane & 15) == 0) {
      int m_glob = base + r + hiOff;
      if (m_glob < A_TOT) {
        best_iou[(size_t)b * A_TOT + m_glob] = v;
        best_idx[(size_t)b * A_TOT + m_glob] = nn;
      }
    }
  }
}

// ---------------------------------------------------------------------------
// Kernel 2: per batch — decode best anchor per GT and replay the scatter in
// n-order (last write wins, matching index_put); also compute any_valid.
// ---------------------------------------------------------------------------
__global__ void scatter_kernel(const unsigned long long* __restrict__ packed,
                               const int* __restrict__ mask,
                               float* __restrict__ best_iou,
                               int* __restrict__ best_idx,
                               int* __restrict__ anyv)
{
  int b = blockIdx.x;
  if (threadIdx.x != 0) return;
  int av = 0;
  for (int n = 0; n < N_GT; ++n) av |= (mask[b * N_GT + n] != 0) ? 1 : 0;
  anyv[b] = av;
  for (int n = 0; n < N_GT; ++n) {
    unsigned long long key = packed[(size_t)b * N_GT + n];
    unsigned m = 0xFFFFFFFFu - (unsigned)(key & 0xFFFFFFFFull);
    if (m < A_TOT) {
      best_idx[(size_t)b * A_TOT + m] = n;
      best_iou[(size_t)b * A_TOT + m] = 2.0f;
    }
  }
}

// ---------------------------------------------------------------------------
// Kernel 3: encode deltas, labels, pos_mask.
// d_out layout: [B*A*4 encoded f32][B*A labels-as-f32][B*A pos_mask-as-f32]
// ---------------------------------------------------------------------------
__global__ void encode_kernel(const float* __restrict__ gt,
                              const int* __restrict__ labels,
                              const float* __restrict__ anchors,
                              const float* __restrict__ best_iou,
                              const int* __restrict__ best_idx,
                              const int* __restrict__ anyv,
                              float* __restrict__ out)
{
  int i = blockIdx.x * blockDim.x + threadIdx.x;
  if (i >= B_SZ * A_TOT) return;
  int b = i / A_TOT, a = i % A_TOT;

  float acx = anchors[a * 4 + 0], acy = anchors[a * 4 + 1];
  float aw  = anchors[a * 4 + 2], ah  = anchors[a * 4 + 3];

  int j = best_idx[i];
  const float* g = gt + ((size_t)b * N_GT + j) * 4;
  float ex = (g[0] - acx) / aw;
  float ey = (g[1] - acy) / ah;
  float ew = __logf((g[2] + EPS_F) / (aw + EPS_F));
  float eh = __logf((g[3] + EPS_F) / (ah + EPS_F));

  float biou = best_iou[i];
  int pos = (biou > 0.5f) ? 1 : 0;
  int lab = pos ? labels[b * N_GT + j] : 0;

  if (!anyv[b]) { ex = 0.f; ey = 0.f; ew = 0.f; eh = 0.f; lab = 0; }

  out[(size_t)i * 4 + 0] = ex;
  out[(size_t)i * 4 + 1] = ey;
  out[(size_t)i * 4 + 2] = ew;
  out[(size_t)i * 4 + 3] = eh;
  out[(size_t)B_SZ * A_TOT * 4 + i] = (float)lab;
  out[(size_t)B_SZ * A_TOT * 5 + i] = pos ? 1.0f : 0.0f;
}

// ---------------------------------------------------------------------------
extern "C" void kernel_launch(void* const* d_in, const int* in_sizes, int n_in,
                              void* d_out, int out_size, void* d_ws, size_t ws_size,
                              hipStream_t stream) {
  (void)in_sizes; (void)n_in; (void)out_size; (void)ws_size;
  const float* gt_boxes  = (const float*)d_in[0];   // [B, N, 4] f32
  const int*   gt_labels = (const int*)d_in[1];     // [B, N]    i32
  const int*   mask      = (const int*)d_in[2];     // [B, N]    bool->int
  const float* anchors   = (const float*)d_in[3];   // [A, 4]    f32 cxcywh
  float* out = (float*)d_out;

  char* ws = (char*)d_ws;
  float* best_iou = (float*)ws;                                    // B*A f32
  int*   best_idx = (int*)(ws + (size_t)B_SZ * A_TOT * 4);         // B*A i32
  unsigned long long* packed =
      (unsigned long long*)(ws + (size_t)2 * B_SZ * A_TOT * 4);    // B*N u64
  int* anyv = (int*)(ws + (size_t)2 * B_SZ * A_TOT * 4 +
                     (size_t)B_SZ * N_GT * 8);                     // B i32

  init_packed_kernel<<<(B_SZ * N_GT + 255) / 256, 256, 0, stream>>>(
      packed, B_SZ * N_GT);
  iou_tile_kernel<<<B_SZ * A_TILES, 32, 0, stream>>>(
      gt_boxes, anchors, best_iou, best_idx, packed);
  scatter_kernel<<<B_SZ, 32, 0, stream>>>(
      packed, mask, best_iou, best_idx, anyv);
  encode_kernel<<<(B_SZ * A_TOT + 255) / 256, 256, 0, stream>>>(
      gt_boxes, gt_labels, anchors, best_iou, best_idx, anyv, out);
}